// ProductionStorage_88381837017300
// MI455X (gfx1250) — compile-verified
//
#include <hip/hip_runtime.h>
#include <cstdint>

// ---------------------------------------------------------------------------
// GR4J-style production storage: elementwise prep + strictly serial 2M-step
// scan (one wave32, readlane broadcast) + WMMA-assisted column reductions +
// in-place normalize.  MI455X / gfx1250.
// ---------------------------------------------------------------------------

typedef float v2f __attribute__((ext_vector_type(2)));
typedef float v8f __attribute__((ext_vector_type(8)));

#if defined(__AMDGCN__)
// Single-instruction xor-shuffle via ds_swizzle_b32 (group-of-32 mode):
// offset = xor_mask<<10 | and_mask(0x1F).  Avoids the ds_bpermute + address
// arithmetic the generic __shfl_xor lowers to.
template <int PAT>
__device__ __forceinline__ float swz_xor(float v) {
  return __uint_as_float(
      (unsigned)__builtin_amdgcn_ds_swizzle((int)__float_as_uint(v), PAT));
}
#endif

// Cross-lane sum over a wave32.  First fold 32 -> 16 with
// V_WMMA_F32_16X16X4_F32 (A = ones 16x4, B = {v, 0}):
//   C[m,n] = sum_k A[m,k]*B[k,n] = v_n + v_{n+16}   (every B/C layout has N = lane%16)
// then finish with ds_swizzle xor folds.  EXEC must be all ones at call site.
__device__ __forceinline__ float wave_sum32(float v) {
#if defined(__gfx1250__) && __has_builtin(__builtin_amdgcn_wmma_f32_16x16x4_f32)
  v2f a; a[0] = 1.0f; a[1] = 1.0f;     // A matrix: all ones
  v2f b; b[0] = v;    b[1] = 0.0f;     // B matrix: lane value in first slot
  v8f c = {};
  c = __builtin_amdgcn_wmma_f32_16x16x4_f32(false, a, false, b,
                                            (short)0, c, false, false);
  float r = c[0];                      // = v[lane%16] + v[lane%16 + 16]
  r += swz_xor<(8  << 10) | 0x1F>(r);  // SWAPX8
  r += swz_xor<(4  << 10) | 0x1F>(r);  // SWAPX4
  r += swz_xor<(2  << 10) | 0x1F>(r);  // SWAPX2
  r += swz_xor<(1  << 10) | 0x1F>(r);  // SWAPX1
  return r;
#elif defined(__AMDGCN__)
  float r = v + swz_xor<(16 << 10) | 0x1F>(v);
  r += swz_xor<(8 << 10) | 0x1F>(r);
  r += swz_xor<(4 << 10) | 0x1F>(r);
  r += swz_xor<(2 << 10) | 0x1F>(r);
  r += swz_xor<(1 << 10) | 0x1F>(r);
  return r;
#else
  float r = v;
  for (int m = 16; m >= 1; m >>= 1) r += __shfl_xor(r, m, 32);
  return r;
#endif
}

// ---------------------------------------------------------------------------
// Kernel 1: the serial scan.  One wave32.  Lanes cooperatively load / prep 32
// timesteps, then 32 fully-unrolled serial steps with v_readlane broadcast of
// lane j's tanh values.  All lanes redundantly carry the uniform state s;
// lane j captures step j's outputs.  global_prefetch_b8 hides HBM latency.
// Serial chain is latency-minimized: hoisted 1/x1, v_rcp-based divides for
// denominators ~= 1, rsq+sqrt for w^-0.25 (w >= 1).  The recurrence is
// contracting, so ~1-ulp per-step errors stay bounded (~1e-6 rel).
// ---------------------------------------------------------------------------
__global__ __launch_bounds__(32) void ps_scan_kernel(
    const float2* __restrict__ x, const float* __restrict__ px1,
    float* __restrict__ out4, float* __restrict__ sstore, int T) {
  const int lane = threadIdx.x;
  const float x1     = *px1;
  const float inv_x1 = 1.0f / x1;                       // one exact divide
  const float cu     = 0.44444444444444444444f * inv_x1; // fp32(4/9)/x1
  float s = 0.0f;

  for (int base = 0; base < T; base += 32) {
    // speculative prefetch 4 chunks (1 KB) ahead -> global_prefetch_b8
    __builtin_prefetch(&x[base + 128 + lane], 0, 0);

    float2 xv = x[base + lane];
    float P = xv.x, E = xv.y;
    float d  = P - E;
    float pn = fmaxf(d, 0.0f);
    float en = fmaxf(-d, 0.0f);
    float tp = tanhf(pn * inv_x1);
    float tn = tanhf(en * inv_x1);

    { float2 pe; pe.x = pn; pe.y = en;      // raw p_n, e_n (normalized later)
      *reinterpret_cast<float2*>(&out4[(size_t)(base + lane) * 4]) = pe; }

    float my_ps = 0.0f, my_perc = 0.0f, my_s = 0.0f;
#pragma unroll
    for (int j = 0; j < 32; ++j) {
      // broadcast step-j operands to SGPRs (immediate-lane readlane)
      float tpj = __uint_as_float(__builtin_amdgcn_readlane(__float_as_uint(tp), j));
      float tnj = __uint_as_float(__builtin_amdgcn_readlane(__float_as_uint(tn), j));

      float r  = s * inv_x1;
      float ps = __fdividef((x1 * (1.0f - r * r)) * tpj, 1.0f + r * tpj);
      float es = __fdividef((s * (2.0f - r)) * tnj, 1.0f + (1.0f - r) * tnj);
      s = s + ps - es;
      float u  = cu * s;
      float u2 = u * u;
      float w  = 1.0f + u2 * u2;
      float wm = sqrtf(rsqrtf(w));                // w^(-0.25), w >= 1
      float perc = s * (1.0f - wm);
      s = s - perc;
      if (lane == j) { my_ps = ps; my_perc = perc; my_s = s; }
    }

    { float2 pp; pp.x = my_ps; pp.y = my_perc;
      *reinterpret_cast<float2*>(&out4[(size_t)(base + lane) * 4 + 2]) = pp; }
    sstore[base + lane] = my_s;
  }
}

// ---------------------------------------------------------------------------
// Kernel 2: per-block partial sum / sum-of-squares of the 4 output columns.
// Deterministic (no float atomics): block b writes partials[b*8 .. b*8+7].
// ---------------------------------------------------------------------------
__global__ __launch_bounds__(256) void reduce_kernel(
    const float* __restrict__ out4, float* __restrict__ partials, int T) {
  const int tid   = threadIdx.x;
  const int gsize = gridDim.x * blockDim.x;
  int gidx = blockIdx.x * blockDim.x + tid;

  float acc[8] = {0, 0, 0, 0, 0, 0, 0, 0};
  const float4* o4 = reinterpret_cast<const float4*>(out4);
  for (int i = gidx; i < T; i += gsize) {
    float4 v = o4[i];
    acc[0] += v.x;       acc[1] += v.y;       acc[2] += v.z;       acc[3] += v.w;
    acc[4] += v.x * v.x; acc[5] += v.y * v.y; acc[6] += v.z * v.z; acc[7] += v.w * v.w;
  }
#pragma unroll
  for (int k = 0; k < 8; ++k) acc[k] = wave_sum32(acc[k]);   // WMMA fold inside

  __shared__ float sm[8][8];
  const int wid = tid >> 5, lane = tid & 31;
  if (lane == 0) {
#pragma unroll
    for (int k = 0; k < 8; ++k) sm[wid][k] = acc[k];
  }
  __syncthreads();
  if (tid < 8) {
    float ssum = 0.0f;
#pragma unroll
    for (int w = 0; w < 8; ++w) ssum += sm[w][tid];
    partials[(size_t)blockIdx.x * 8 + tid] = ssum;
  }
}

// ---------------------------------------------------------------------------
// Kernel 3: fold 256 block partials -> mu[4], 1/sigma[4]  (population std).
// ---------------------------------------------------------------------------
__global__ __launch_bounds__(256) void finalize_kernel(
    const float* __restrict__ partials, float* __restrict__ stats,
    int nblk, int T) {
  const int tid = threadIdx.x;
  float acc[8] = {0, 0, 0, 0, 0, 0, 0, 0};
  for (int b = tid; b < nblk; b += 256) {
#pragma unroll
    for (int k = 0; k < 8; ++k) acc[k] += partials[(size_t)b * 8 + k];
  }
#pragma unroll
  for (int k = 0; k < 8; ++k) acc[k] = wave_sum32(acc[k]);

  __shared__ float sm[8][8];
  __shared__ float tot[8];
  const int wid = tid >> 5, lane = tid & 31;
  if (lane == 0) {
#pragma unroll
    for (int k = 0; k < 8; ++k) sm[wid][k] = acc[k];
  }
  __syncthreads();
  if (tid < 8) {
    float ssum = 0.0f;
#pragma unroll
    for (int w = 0; w < 8; ++w) ssum += sm[w][tid];
    tot[tid] = ssum;
  }
  __syncthreads();
  if (tid < 4) {
    const float invT = 1.0f / (float)T;
    float mu  = tot[tid] * invT;
    float var = fmaxf(tot[tid + 4] * invT - mu * mu, 0.0f);
    stats[tid]     = mu;
    stats[tid + 4] = 1.0f / sqrtf(var);
  }
}

// ---------------------------------------------------------------------------
// Kernel 4: in-place (v - mu) * (1/sigma) over float4 rows (b128 coalesced).
// ---------------------------------------------------------------------------
__global__ __launch_bounds__(256) void normalize_kernel(
    float* __restrict__ out4, const float* __restrict__ stats, int T) {
  int i = blockIdx.x * blockDim.x + threadIdx.x;
  if (i >= T) return;
  float4* o4 = reinterpret_cast<float4*>(out4);
  float4 v = o4[i];
  v.x = (v.x - stats[0]) * stats[4];
  v.y = (v.y - stats[1]) * stats[5];
  v.z = (v.z - stats[2]) * stats[6];
  v.w = (v.w - stats[3]) * stats[7];
  o4[i] = v;
}

// ---------------------------------------------------------------------------
extern "C" void kernel_launch(void* const* d_in, const int* in_sizes, int n_in,
                              void* d_out, int out_size, void* d_ws, size_t ws_size,
                              hipStream_t stream) {
  const float2* x   = (const float2*)d_in[0];
  const float*  px1 = (const float*)d_in[1];
  const int T = in_sizes[0] / 2;

  float* out4     = (float*)d_out;                // (T,4) row-major
  float* sstore   = out4 + (size_t)T * 4;         // (T,1)
  float* partials = (float*)d_ws;                 // 256 blocks * 8 floats
  float* stats    = partials + 256 * 8;           // mu[4], rsigma[4]

  ps_scan_kernel<<<1, 32, 0, stream>>>(x, px1, out4, sstore, T);
  reduce_kernel<<<256, 256, 0, stream>>>(out4, partials, T);
  finalize_kernel<<<1, 256, 0, stream>>>(partials, stats, 256, T);
  normalize_kernel<<<(T + 255) / 256, 256, 0, stream>>>(out4, stats, T);
}